// LGNRecModel_63857573757118
// MI455X (gfx1250) — compile-verified
//
#include <hip/hip_runtime.h>
#include <math.h>

#define DIM 64

typedef __attribute__((ext_vector_type(2))) float v2f;
typedef __attribute__((ext_vector_type(4))) float v4f;
typedef __attribute__((ext_vector_type(8))) float v8f;

// ---------------- degree / norm ----------------

__global__ void lgn_zero_f32(float* __restrict__ p, int n) {
    int i = blockIdx.x * blockDim.x + threadIdx.x;
    if (i < n) p[i] = 0.0f;
}

__global__ void lgn_degree(const int* __restrict__ hs, const int* __restrict__ ts,
                           float* __restrict__ deg, int nE) {
    int e = blockIdx.x * blockDim.x + threadIdx.x;
    if (e < nE) {
        atomicAdd(&deg[hs[e]], 1.0f);
        atomicAdd(&deg[ts[e]], 1.0f);
    }
}

__global__ void lgn_norm(float* __restrict__ deg_norm, int n) {
    int i = blockIdx.x * blockDim.x + threadIdx.x;
    if (i < n) deg_norm[i] = 1.0f / sqrtf(fmaxf(deg_norm[i], 1.0f));
}

// ---------------- propagation ----------------

// gcn = entity_emb (layer-0 accumulation); x = entity_emb * norm; agg = 0
__global__ void lgn_init(const float* __restrict__ ent, const float* __restrict__ norm,
                         float* __restrict__ gcn, float* __restrict__ x,
                         float* __restrict__ agg, int n4) {
    int i = blockIdx.x * blockDim.x + threadIdx.x;
    if (i >= n4) return;
    int row = i >> 4;                 // 16 float4 per 64-dim row
    float nr = norm[row];
    v4f e = ((const v4f*)ent)[i];
    ((v4f*)gcn)[i] = e;
    ((v4f*)x)[i]   = e * nr;
    v4f z = {0.f, 0.f, 0.f, 0.f};
    ((v4f*)agg)[i] = z;
}

// agg[ts[e]] += x[hs[e]]  (16 lanes per edge, float4 per lane, f32 atomics -> L2)
__global__ void lgn_scatter(const float* __restrict__ x, const int* __restrict__ hs,
                            const int* __restrict__ ts, float* __restrict__ agg, int nE) {
    int tid = blockIdx.x * blockDim.x + threadIdx.x;
    int e = tid >> 4;
    if (e >= nE) return;
    int sub = (tid & 15) << 2;        // 0,4,...,60
    int h = hs[e];
    int t = ts[e];
    v4f v = *(const v4f*)(x + (size_t)h * DIM + sub);
    float* dst = agg + (size_t)t * DIM + sub;
    atomicAdd(dst + 0, v[0]);
    atomicAdd(dst + 1, v[1]);
    atomicAdd(dst + 2, v[2]);
    atomicAdd(dst + 3, v[3]);
}

// new = agg*norm; gcn += new; x = new*norm; agg = 0
__global__ void lgn_finish(float* __restrict__ agg, const float* __restrict__ norm,
                           float* __restrict__ gcn, float* __restrict__ x, int n4) {
    int i = blockIdx.x * blockDim.x + threadIdx.x;
    if (i >= n4) return;
    int row = i >> 4;
    float nr = norm[row];
    v4f a  = ((const v4f*)agg)[i];
    v4f nv = a * nr;
    v4f g  = ((const v4f*)gcn)[i];
    ((v4f*)gcn)[i] = g + nv;
    ((v4f*)x)[i]   = nv * nr;
    v4f z = {0.f, 0.f, 0.f, 0.f};
    ((v4f*)agg)[i] = z;
}

// ---------------- scoring: WMMA f32 16x16x4 dual dot products ----------------
// One wave handles 16 samples. D = U(16x64) x P^T(64x16) via 16 chained K=4 WMMAs;
// diagonal of D gives per-sample dot products. EXEC is all-ones (no divergence
// before the WMMAs), as required by the WMMA ISA restriction.
__global__ void lgn_score(const float* __restrict__ user_emb, const float* __restrict__ gcn,
                          const int* __restrict__ users, const int* __restrict__ pos_items,
                          const int* __restrict__ neg_items, float* __restrict__ out) {
    int tid  = blockIdx.x * blockDim.x + threadIdx.x;
    int wave = tid >> 5;
    int lane = tid & 31;
    int r    = lane & 15;             // A row / B column (sample within tile)
    int off  = (lane >> 4) << 1;      // K sub-offset: 0 (lanes 0-15) or 2 (lanes 16-31)
    int s    = wave * 16 + r;

    const float* U  = user_emb + (size_t)users[s]     * DIM;
    const float* P  = gcn      + (size_t)pos_items[s] * DIM;
    const float* Ng = gcn      + (size_t)neg_items[s] * DIM;

    v8f cp = {0.f, 0.f, 0.f, 0.f, 0.f, 0.f, 0.f, 0.f};
    v8f cn = {0.f, 0.f, 0.f, 0.f, 0.f, 0.f, 0.f, 0.f};

#pragma unroll
    for (int kk = 0; kk < DIM / 4; ++kk) {
        int k = kk * 4 + off;
        v2f a  = *(const v2f*)(U  + k);   // A[m=r][k], A[m=r][k+1]
        v2f bp = *(const v2f*)(P  + k);   // B[k][n=r] = P[n=r][k]
        v2f bn = *(const v2f*)(Ng + k);
        cp = __builtin_amdgcn_wmma_f32_16x16x4_f32(false, a, false, bp, (short)0, cp, false, false);
        cn = __builtin_amdgcn_wmma_f32_16x16x4_f32(false, a, false, bn, (short)0, cn, false, false);
    }

    // Diagonal extraction from the 16x16 f32 C/D layout:
    //   M<8:  lane = N,      vgpr = M      -> lanes 0..7,  c[lane]
    //   M>=8: lane = N + 16, vgpr = M - 8  -> lanes 24..31, c[lane-24]
    int outIdx = -1;
    float pos = 0.f, neg = 0.f;
    if (lane < 8) {
        pos = cp[lane];      neg = cn[lane];      outIdx = wave * 16 + lane;
    } else if (lane >= 24) {
        pos = cp[lane - 24]; neg = cn[lane - 24]; outIdx = wave * 16 + (lane - 16);
    }
    if (outIdx >= 0) {
        float xv = (neg - pos) * 0.25f;   // item_embedding = gcn / 4 folded into scores
        // stable softplus
        out[outIdx] = fmaxf(xv, 0.f) + log1pf(expf(-fabsf(xv)));
    }
}

// ---------------- host launch ----------------

extern "C" void kernel_launch(void* const* d_in, const int* in_sizes, int n_in,
                              void* d_out, int out_size, void* d_ws, size_t ws_size,
                              hipStream_t stream) {
    const float* entity_emb = (const float*)d_in[0];
    const float* user_emb   = (const float*)d_in[1];
    const int*   users      = (const int*)d_in[2];
    const int*   pos_items  = (const int*)d_in[3];
    const int*   neg_items  = (const int*)d_in[4];
    const int*   hs         = (const int*)d_in[5];
    const int*   ts         = (const int*)d_in[6];
    float*       out        = (float*)d_out;

    const int nEnt     = in_sizes[0] / DIM;   // 100000
    const int nEdges   = in_sizes[5];         // 1000000
    const int nSamples = in_sizes[2];         // 4096

    // workspace layout (floats)
    float* ws   = (float*)d_ws;
    size_t o    = 0;
    float* norm = ws + o; o += ((size_t)nEnt + 63) & ~(size_t)63;
    float* x    = ws + o; o += (size_t)nEnt * DIM;
    float* agg  = ws + o; o += (size_t)nEnt * DIM;
    float* gcn  = ws + o; o += (size_t)nEnt * DIM;
    (void)ws_size; (void)n_in; (void)out_size;

    const int B = 256;
    const int n4 = nEnt * (DIM / 4);          // float4 elements per emb buffer

    // degrees -> norm (norm buffer doubles as degree accumulator)
    lgn_zero_f32<<<(nEnt + B - 1) / B, B, 0, stream>>>(norm, nEnt);
    lgn_degree  <<<(nEdges + B - 1) / B, B, 0, stream>>>(hs, ts, norm, nEdges);
    lgn_norm    <<<(nEnt + B - 1) / B, B, 0, stream>>>(norm, nEnt);

    // init: gcn = e0, x = e0*norm, agg = 0
    lgn_init<<<(n4 + B - 1) / B, B, 0, stream>>>(entity_emb, norm, gcn, x, agg, n4);

    // 3 propagation layers
    for (int layer = 0; layer < 3; ++layer) {
        int scatterThreads = nEdges * 16;
        lgn_scatter<<<(scatterThreads + B - 1) / B, B, 0, stream>>>(x, hs, ts, agg, nEdges);
        lgn_finish <<<(n4 + B - 1) / B, B, 0, stream>>>(agg, norm, gcn, x, n4);
    }

    // scoring: 16 samples per wave, 32 threads per wave
    int nWaves   = nSamples / 16;             // 256
    int nThreads = nWaves * 32;               // 8192
    lgn_score<<<nThreads / B, B, 0, stream>>>(user_emb, gcn, users, pos_items, neg_items, out);
}